// Nalui2Layer_55594056679945
// MI455X (gfx1250) — compile-verified
//
#include <hip/hip_runtime.h>
#include <math.h>

// NALU-i2 layer, B=1024, IN=OUT=512, fp32 throughout.
//
// Decomposition (see analysis): the whole layer is 4 GEMMs sharing A-side
// K=512 plus an elementwise epilogue:
//   a1     = X    @ W1                 W1[i][o] = tanh(w1h)*sigmoid(m1h)
//   logm1  = logX @ W2                 W2[i][o] = tanh(w2h)*sigmoid(m2h)
//   log|ms|= Neg  @ L                  L[i][o]  = log|1-2|W2[o][i]||   (transposed W2 indexing!)
//   parity = Neg  @ S                  S[i][o]  = (1-2|W2[o][i]|) < 0
// Neg[b][i] = (x<0).  (sign(0)=0 case ignored: measure-zero for normal inputs.)
// Epilogue: out = g*a1 + (1-g)*exp(min(logm1,20))*clip((-1)^parity*exp(log|ms|),-1,1)
//
// fp32 WMMA V_WMMA_F32_16X16X4_F32: one wave computes one 16x16 output tile,
// accumulating 4 matmuls over K=512 in steps of 4. B matrices are stored
// transposed ([out][in]) in workspace so every WMMA operand is one contiguous
// float2 load per lane.

#define B_   1024
#define IN_  512
#define OUT_ 512

typedef __attribute__((ext_vector_type(2))) float v2f;
typedef __attribute__((ext_vector_type(8))) float v8f;

__device__ __forceinline__ float sigm(float x) { return 1.0f / (1.0f + __expf(-x)); }

// ---------------------------------------------------------------------------
// Prep 1: transformed weights. Thread j = r*512 + c over w_hat flat storage.
//   W1t[c*512+r] = W1[r][c]   (transpose -> [out][in])
//   W2t[c*512+r] = W2[r][c]   (transpose -> [out][in])
//   Lt [j]       = log|1-2|W2[r][c]||   ([o][i] layout with o=r, per the
//                                        reference's tile/reshape indexing)
//   St [j]       = (1-2|W2[r][c]|) < 0 ? 1 : 0
// ---------------------------------------------------------------------------
__global__ void prep_weights(const float* __restrict__ w1h, const float* __restrict__ m1h,
                             const float* __restrict__ w2h, const float* __restrict__ m2h,
                             float* __restrict__ W1t, float* __restrict__ W2t,
                             float* __restrict__ Lt,  float* __restrict__ St) {
    int j = blockIdx.x * blockDim.x + threadIdx.x;
    if (j >= IN_ * OUT_) return;
    int r = j >> 9;
    int c = j & 511;
    float v1 = tanhf(w1h[j]) * sigm(m1h[j]);
    float v2 = tanhf(w2h[j]) * sigm(m2h[j]);
    W1t[c * IN_ + r] = v1;
    W2t[c * IN_ + r] = v2;
    float t = 1.0f - 2.0f * fabsf(v2);
    Lt[j] = __logf(fmaxf(fabsf(t), 1e-30f)); // clamp avoids -inf -> 0*inf NaN in matmul
    St[j] = (t < 0.0f) ? 1.0f : 0.0f;
}

// ---------------------------------------------------------------------------
// Prep 2: per-element input transforms.
// ---------------------------------------------------------------------------
__global__ void prep_x(const float* __restrict__ X,
                       float* __restrict__ LX, float* __restrict__ NG) {
    int j = blockIdx.x * blockDim.x + threadIdx.x;
    if (j >= B_ * IN_) return;
    float x = X[j];
    LX[j] = __logf(fmaxf(fabsf(x), 1e-7f));
    NG[j] = (x < 0.0f) ? 1.0f : 0.0f;
}

// ---------------------------------------------------------------------------
// Fused 4-way GEMM + epilogue. One wave -> one 16x16 tile of the output.
// 2048 tiles total; 8 waves (256 threads) per block; 256 blocks.
//
// f32 WMMA operand layouts (CDNA5 ISA 7.12.2):
//   A 16x4 : lanes 0-15 hold {K=0,K=1} of rows M=lane; lanes 16-31 hold {K=2,K=3}
//   B 4x16 : lanes 0-15 hold {K=0,K=1} of col N=lane;  lanes 16-31 hold {K=2,K=3}
//   C 16x16: VGPR r holds row M = r + 8*(lane>=16), col N = lane%16
// With B stored [out][in], both A and B loads are contiguous float2 per lane.
// ---------------------------------------------------------------------------
__global__ __launch_bounds__(256)
void nalu_fused_gemm(const float* __restrict__ X,   const float* __restrict__ LX,
                     const float* __restrict__ NG,
                     const float* __restrict__ W1t, const float* __restrict__ W2t,
                     const float* __restrict__ Lt,  const float* __restrict__ St,
                     const float* __restrict__ G1,  float* __restrict__ out) {
    const int wave = threadIdx.x >> 5;
    const int lane = threadIdx.x & 31;
    const int tile = blockIdx.x * 8 + wave;      // 0..2047
    const int tm   = tile & 63;                  // 64 row tiles (batch)
    const int tn   = tile >> 6;                  // 32 col tiles (out)
    const int row0 = tm * 16;
    const int col0 = tn * 16;
    const int lo   = lane & 15;
    const int hi   = lane >> 4;                  // 0: K=0,1  1: K=2,3

    const int arow = row0 + lo;                  // A-side row this lane feeds
    const int n    = col0 + lo;                  // B-side column this lane feeds
    const float* pX  = X   + arow * IN_;
    const float* pLX = LX  + arow * IN_;
    const float* pNG = NG  + arow * IN_;
    const float* pW1 = W1t + n * IN_;
    const float* pW2 = W2t + n * IN_;
    const float* pLt = Lt  + n * IN_;
    const float* pSt = St  + n * IN_;

    v8f c_a = {};   // a1 accumulator
    v8f c_m = {};   // log-mult accumulator
    v8f c_s = {};   // log|ms1| accumulator
    v8f c_p = {};   // negative-factor parity count

#pragma unroll 4
    for (int kk = 0; kk < IN_; kk += 4) {
        const int ko = kk + 2 * hi;
        v2f ax = *(const v2f*)(pX  + ko);
        v2f al = *(const v2f*)(pLX + ko);
        v2f an = *(const v2f*)(pNG + ko);
        v2f b1 = *(const v2f*)(pW1 + ko);
        v2f b2 = *(const v2f*)(pW2 + ko);
        v2f bl = *(const v2f*)(pLt + ko);
        v2f bs = *(const v2f*)(pSt + ko);
        c_a = __builtin_amdgcn_wmma_f32_16x16x4_f32(false, ax, false, b1, (short)0, c_a, false, false);
        c_m = __builtin_amdgcn_wmma_f32_16x16x4_f32(false, al, false, b2, (short)0, c_m, false, false);
        c_s = __builtin_amdgcn_wmma_f32_16x16x4_f32(false, an, false, bl, (short)0, c_s, false, false);
        c_p = __builtin_amdgcn_wmma_f32_16x16x4_f32(false, an, false, bs, (short)0, c_p, false, false);
    }

    const float g = sigm(G1[n]);
#pragma unroll
    for (int r = 0; r < 8; ++r) {
        const int m  = row0 + r + 8 * hi;
        const float a1  = c_a[r];
        const float m1  = __expf(fminf(c_m[r], 20.0f));
        const float mag = __expf(c_s[r]);
        const int   par = ((int)(c_p[r] + 0.5f)) & 1;   // exact integer count in fp32
        float ms1 = par ? -mag : mag;
        ms1 = fminf(fmaxf(ms1, -1.0f), 1.0f);
        out[m * OUT_ + n] = g * a1 + (1.0f - g) * m1 * ms1;
    }
}

// ---------------------------------------------------------------------------
// Launch. Workspace layout (floats): W1t | W2t | Lt | St | logX | Neg
//   4 * 262144 + 2 * 524288 = 2097152 floats = 8 MB.
// ---------------------------------------------------------------------------
extern "C" void kernel_launch(void* const* d_in, const int* in_sizes, int n_in,
                              void* d_out, int out_size, void* d_ws, size_t ws_size,
                              hipStream_t stream) {
    const float* X   = (const float*)d_in[0];
    const float* w1h = (const float*)d_in[1];
    const float* m1h = (const float*)d_in[2];
    const float* w2h = (const float*)d_in[3];
    const float* m2h = (const float*)d_in[4];
    const float* G1  = (const float*)d_in[5];
    float* outp = (float*)d_out;

    float* ws  = (float*)d_ws;
    float* W1t = ws;
    float* W2t = W1t + IN_ * OUT_;
    float* Lt  = W2t + IN_ * OUT_;
    float* St  = Lt  + IN_ * OUT_;
    float* LX  = St  + IN_ * OUT_;
    float* NG  = LX  + B_ * IN_;

    prep_weights<<<(IN_ * OUT_) / 256, 256, 0, stream>>>(w1h, m1h, w2h, m2h, W1t, W2t, Lt, St);
    prep_x<<<(B_ * IN_) / 256, 256, 0, stream>>>(X, LX, NG);
    nalu_fused_gemm<<<256, 256, 0, stream>>>(X, LX, NG, W1t, W2t, Lt, St, G1, outp);
}